// PointTransformer_59579786330698
// MI455X (gfx1250) — compile-verified
//
#include <hip/hip_runtime.h>
#include <math.h>

#define B_  16
#define T_  256
#define CTOK 256
#define INNER 64
#define TH_ 32
#define TPR 10
#define EPS 1e-5f

typedef __attribute__((ext_vector_type(2))) float v2f;
typedef __attribute__((ext_vector_type(8))) float v8f;

// ---- CDNA5 f32 WMMA: D(16x16,f32) = A(16x4,f32) x B(4x16,f32) + C ----
__device__ __forceinline__ v8f wmma4(v2f a, v2f b, v8f c) {
  return __builtin_amdgcn_wmma_f32_16x16x4_f32(
      /*neg_a=*/false, a, /*neg_b=*/false, b,
      /*c_mod=*/(short)0, c, /*reuse_a=*/false, /*reuse_b=*/false);
}

// A fragment (16x4) from row-major global: lanes 0-15 -> rows, halves -> K pairs
__device__ __forceinline__ v2f a_frag_g(const float* __restrict__ A, int lda,
                                        int row0, int k0, int lane) {
  int row = row0 + (lane & 15);
  int kb  = k0 + ((lane >> 4) << 1);
  const float* p = A + (size_t)row * lda + kb;
  v2f a; a.x = p[0]; a.y = p[1]; return a;
}
// B fragment (4x16) from row-major global
__device__ __forceinline__ v2f b_frag_g(const float* __restrict__ Bm, int ldb,
                                        int k0, int col0, int lane) {
  int col = col0 + (lane & 15);
  int kb  = k0 + ((lane >> 4) << 1);
  v2f b; b.x = Bm[(size_t)kb * ldb + col]; b.y = Bm[(size_t)(kb + 1) * ldb + col];
  return b;
}
// A fragment from LDS tile (row-major, padded stride)
__device__ __forceinline__ v2f a_frag_s(const float* S, int stride, int k0, int lane) {
  int row = lane & 15;
  int kb  = k0 + ((lane >> 4) << 1);
  const float* p = S + row * stride + kb;
  v2f a; a.x = p[0]; a.y = p[1]; return a;
}

// ============ fc1: y = BN(x@W1+b1); skip=y; ya=relu(y). One wave / 16x16 tile.
__global__ __launch_bounds__(32) void fc1_kernel(
    const float* __restrict__ x, const float* __restrict__ W1,
    const float* __restrict__ b1, const float* __restrict__ g,
    const float* __restrict__ bb, const float* __restrict__ mm,
    const float* __restrict__ vv, float* __restrict__ skip,
    float* __restrict__ ya) {
  const int lane = threadIdx.x;
  const int nt = blockIdx.x & 15, mt = blockIdx.x >> 4;
  const int row0 = mt << 4, col0 = nt << 4;
  v8f c = {};
#pragma unroll 4
  for (int k = 0; k < CTOK; k += 4) {
    v2f a = a_frag_g(x, CTOK, row0, k, lane);
    v2f b = b_frag_g(W1, CTOK, k, col0, lane);
    c = wmma4(a, b, c);
  }
  const int col = col0 + (lane & 15);
  const float sc = g[col] * rsqrtf(vv[col] + EPS);
  const float tt = (b1[col] - mm[col]) * sc + bb[col];
  const int rb = row0 + ((lane >> 4) << 3);
#pragma unroll
  for (int v = 0; v < 8; ++v) {
    float y = c[v] * sc + tt;
    skip[(size_t)(rb + v) * CTOK + col] = y;
    ya[(size_t)(rb + v) * CTOK + col]   = fmaxf(y, 0.f);
  }
}

// ============ generic C = A@B (no bias). One wave / 16x16 tile.
__global__ __launch_bounds__(32) void gemm_kernel(
    const float* __restrict__ A, int lda, const float* __restrict__ Bm, int ldb,
    float* __restrict__ C, int ldc, int ntiles, int K) {
  const int lane = threadIdx.x;
  const int nt = blockIdx.x % ntiles, mt = blockIdx.x / ntiles;
  const int row0 = mt << 4, col0 = nt << 4;
  v8f c = {};
#pragma unroll 4
  for (int k = 0; k < K; k += 4) {
    v2f a = a_frag_g(A, lda, row0, k, lane);
    v2f b = b_frag_g(Bm, ldb, k, col0, lane);
    c = wmma4(a, b, c);
  }
  const int col = col0 + (lane & 15);
  const int rb = row0 + ((lane >> 4) << 3);
#pragma unroll
  for (int v = 0; v < 8; ++v) C[(size_t)(rb + v) * ldc + col] = c[v];
}

// ============ fused pairwise attention: one workgroup (4 waves) per (b,i)
#define HS 34   // hidden tile stride (16x32); 34r mod 64 distinct, halves disjoint
#define RS 68   // rel/h tile stride (16x64); 4r pattern, halves disjoint

__global__ __launch_bounds__(128) void pt_attn_kernel(
    const float* __restrict__ r, const float* __restrict__ qkv,
    const float* __restrict__ qt, const float* __restrict__ kt,
    const float* __restrict__ Wt1, const float* __restrict__ bt1,
    const float* __restrict__ bntg, const float* __restrict__ bntb,
    const float* __restrict__ bntm, const float* __restrict__ bntv,
    const float* __restrict__ Wt2, const float* __restrict__ bt2,
    const float* __restrict__ Wa1, const float* __restrict__ ba1,
    const float* __restrict__ Wa2, const float* __restrict__ ba2,
    const float* __restrict__ W2, const float* __restrict__ b2,
    const float* __restrict__ bn2g, const float* __restrict__ bn2b,
    const float* __restrict__ bn2m, const float* __restrict__ bn2v,
    const float* __restrict__ skip, float* __restrict__ out) {
  __shared__ float sHid[16 * HS];
  __shared__ float sRel[16 * RS];
  __shared__ float sH[16 * RS];
  __shared__ float sAgg[64];

  const int tid = threadIdx.x;
  const int lane = tid & 31;
  const int wave = tid >> 5;
  const int row_i = blockIdx.x;           // = b*T + i
  const int btok = row_i & ~(T_ - 1);     // = b*T
  const float* rbase = r + (size_t)row_i * T_ * TPR;

  const int col0 = wave << 4;             // this wave's N-slice (0..3)
  const int col  = col0 + (lane & 15);
  const int rb   = (lane >> 4) << 3;      // C-fragment row base
  const int kh   = (lane >> 4) << 1;      // K sub-offset per lane half
  const int ca   = (wave & 1) << 4;       // tpr-hidden slice for stage A
  const int hcol = ca + (lane & 15);

  // ---- loop-invariant weight B-fragments held in VGPRs (L2-hot, loaded once)
  v2f fWt1[3], fWt2[8], fWa1[16], fWa2[16];
#pragma unroll
  for (int s = 0; s < 3; ++s) {
    int kb = s * 4 + kh;
    fWt1[s].x = (kb < TPR)     ? Wt1[kb * TH_ + hcol]       : 0.f;
    fWt1[s].y = (kb + 1 < TPR) ? Wt1[(kb + 1) * TH_ + hcol] : 0.f;
  }
#pragma unroll
  for (int s = 0; s < 8; ++s) {
    int kb = s * 4 + kh;
    fWt2[s].x = Wt2[kb * INNER + col];
    fWt2[s].y = Wt2[(kb + 1) * INNER + col];
  }
#pragma unroll
  for (int s = 0; s < 16; ++s) {
    int kb = s * 4 + kh;
    fWa1[s].x = Wa1[kb * INNER + col];
    fWa1[s].y = Wa1[(kb + 1) * INNER + col];
    fWa2[s].x = Wa2[kb * INNER + col];
    fWa2[s].y = Wa2[(kb + 1) * INNER + col];
  }

  // ---- per-lane scalar constants (fused BN / biases / q~ fold)
  const float scH = bntg[hcol] * rsqrtf(bntv[hcol] + EPS);
  const float ttH = (bt1[hcol] - bntm[hcol]) * scH + bntb[hcol];
  const float btc = bt2[col];
  const float bac = ba2[col];
  const float qb  = qt[(size_t)row_i * INNER + col] + ba1[col];

  float mrun = -3.0e38f, ssum = 0.f, accv = 0.f;

  for (int j0 = 0; j0 < T_; j0 += 16) {
    // prefetch next r tile (640B) while this one computes
    if (j0 + 16 < T_ && tid < 10)
      __builtin_prefetch(rbase + (j0 + 16) * TPR + tid * 16, 0, 3);

    // ---- stage A: hidden = relu(BN(r_tile @ Wt1)) ; K=10 padded to 12
    {
      v8f hc = {};
#pragma unroll
      for (int s = 0; s < 3; ++s) {
        int kb = s * 4 + kh;
        int jrow = j0 + (lane & 15);
        const float* rp = rbase + jrow * TPR + kb;
        v2f a;
        a.x = (kb < TPR) ? rp[0] : 0.f;
        a.y = (kb + 1 < TPR) ? rp[1] : 0.f;
        hc = wmma4(a, fWt1[s], hc);
      }
#pragma unroll
      for (int v = 0; v < 8; ++v)
        sHid[(rb + v) * HS + hcol] = fmaxf(hc[v] * scH + ttH, 0.f);
    }
    __syncthreads();

    // ---- stage B: rel = hidden @ Wt2 + bt2 (K=32); keep C-frag for vv
    float relv[8];
    {
      v8f rc = {};
#pragma unroll
      for (int s = 0; s < 8; ++s) {
        v2f a = a_frag_s(sHid, HS, s * 4, lane);
        rc = wmma4(a, fWt2[s], rc);
      }
#pragma unroll
      for (int v = 0; v < 8; ++v) {
        relv[v] = rc[v] + btc;
        sRel[(rb + v) * RS + col] = relv[v];
      }
    }
    __syncthreads();

    // ---- stage C: h = relu(q~ - k~ + rel@Wa1 + ba1) (K=64)
    {
      v8f hc = {};
#pragma unroll
      for (int s = 0; s < 16; ++s) {
        v2f a = a_frag_s(sRel, RS, s * 4, lane);
        hc = wmma4(a, fWa1[s], hc);
      }
#pragma unroll
      for (int v = 0; v < 8; ++v) {
        int jr = j0 + rb + v;
        float ktv = kt[(size_t)(btok + jr) * INNER + col];
        sH[(rb + v) * RS + col] = fmaxf(hc[v] + qb - ktv, 0.f);
      }
    }
    __syncthreads();

    // ---- stage D: sim = h @ Wa2 + ba2 (K=64)
    float sim[8];
    {
      v8f sc = {};
#pragma unroll
      for (int s = 0; s < 16; ++s) {
        v2f a = a_frag_s(sH, RS, s * 4, lane);
        sc = wmma4(a, fWa2[s], sc);
      }
#pragma unroll
      for (int v = 0; v < 8; ++v) sim[v] = sc[v] + bac;
    }

    // ---- stage E: online softmax over j (per channel), vv = v_j + rel
    {
      float tmax = sim[0];
#pragma unroll
      for (int v = 1; v < 8; ++v) tmax = fmaxf(tmax, sim[v]);
      tmax = fmaxf(tmax, __shfl_xor(tmax, 16, 32));
      float mnew = fmaxf(mrun, tmax);
      float scale = __expf(mrun - mnew);
      float psum = 0.f, vsum = 0.f;
#pragma unroll
      for (int v = 0; v < 8; ++v) {
        float e = __expf(sim[v] - mnew);
        int jr = j0 + rb + v;
        float vj = qkv[(size_t)(btok + jr) * 192 + 128 + col];
        psum += e;
        vsum += e * (vj + relv[v]);
      }
      psum += __shfl_xor(psum, 16, 32);
      vsum += __shfl_xor(vsum, 16, 32);
      ssum = ssum * scale + psum;
      accv = accv * scale + vsum;
      mrun = mnew;
    }
    __syncthreads();  // tiles reused next iteration
  }

  if (lane < 16) sAgg[col] = accv / ssum;
  __syncthreads();

  // ---- fused fc2 + BN + skip + relu (256 outputs, 2 per thread)
  {
    const int n0 = tid << 1;
    float a0 = b2[n0], a1 = b2[n0 + 1];
#pragma unroll 8
    for (int c2 = 0; c2 < 64; ++c2) {
      float gv = sAgg[c2];
      const float* wp = W2 + c2 * CTOK + n0;
      a0 = fmaf(gv, wp[0], a0);
      a1 = fmaf(gv, wp[1], a1);
    }
    size_t ob = (size_t)row_i * CTOK + n0;
    float s0 = bn2g[n0] * rsqrtf(bn2v[n0] + EPS);
    float s1 = bn2g[n0 + 1] * rsqrtf(bn2v[n0 + 1] + EPS);
    float o0 = (a0 - bn2m[n0]) * s0 + bn2b[n0] + skip[ob];
    float o1 = (a1 - bn2m[n0 + 1]) * s1 + bn2b[n0 + 1] + skip[ob + 1];
    out[ob]     = fmaxf(o0, 0.f);
    out[ob + 1] = fmaxf(o1, 0.f);
  }
}

extern "C" void kernel_launch(void* const* d_in, const int* in_sizes, int n_in,
                              void* d_out, int out_size, void* d_ws, size_t ws_size,
                              hipStream_t stream) {
  (void)in_sizes; (void)n_in; (void)out_size; (void)ws_size;
  const float* x    = (const float*)d_in[0];
  const float* r    = (const float*)d_in[1];
  const float* W1   = (const float*)d_in[2];
  const float* b1   = (const float*)d_in[3];
  const float* bn1g = (const float*)d_in[4];
  const float* bn1b = (const float*)d_in[5];
  const float* bn1m = (const float*)d_in[6];
  const float* bn1v = (const float*)d_in[7];
  const float* Wqkv = (const float*)d_in[8];
  const float* Wt1  = (const float*)d_in[9];
  const float* bt1  = (const float*)d_in[10];
  const float* bntg = (const float*)d_in[11];
  const float* bntb = (const float*)d_in[12];
  const float* bntm = (const float*)d_in[13];
  const float* bntv = (const float*)d_in[14];
  const float* Wt2  = (const float*)d_in[15];
  const float* bt2  = (const float*)d_in[16];
  const float* Wa1  = (const float*)d_in[17];
  const float* ba1  = (const float*)d_in[18];
  const float* Wa2  = (const float*)d_in[19];
  const float* ba2  = (const float*)d_in[20];
  const float* W2   = (const float*)d_in[21];
  const float* b2   = (const float*)d_in[22];
  const float* bn2g = (const float*)d_in[23];
  const float* bn2b = (const float*)d_in[24];
  const float* bn2m = (const float*)d_in[25];
  const float* bn2v = (const float*)d_in[26];
  float* outp = (float*)d_out;

  const int NT = B_ * T_;                 // 4096 token rows
  float* ws   = (float*)d_ws;
  float* skip = ws;                       // 4096*256
  float* ya   = skip + (size_t)NT * CTOK; // 4096*256
  float* qkv  = ya + (size_t)NT * CTOK;   // 4096*192
  float* qtb  = qkv + (size_t)NT * 192;   // 4096*64
  float* ktb  = qtb + (size_t)NT * INNER; // 4096*64

  // 1) fc1 + BN (-> skip, ya)
  fc1_kernel<<<(NT / 16) * (CTOK / 16), 32, 0, stream>>>(
      x, W1, b1, bn1g, bn1b, bn1m, bn1v, skip, ya);
  // 2) qkv = ya @ Wqkv   (M=4096, N=192, K=256)
  gemm_kernel<<<(NT / 16) * (192 / 16), 32, 0, stream>>>(
      ya, CTOK, Wqkv, 192, qkv, 192, 192 / 16, CTOK);
  // 3) q~ = q @ Wa1, k~ = k @ Wa1 (fold Wa1 out of pairwise path)
  gemm_kernel<<<(NT / 16) * (INNER / 16), 32, 0, stream>>>(
      qkv, 192, Wa1, INNER, qtb, INNER, INNER / 16, INNER);
  gemm_kernel<<<(NT / 16) * (INNER / 16), 32, 0, stream>>>(
      qkv + INNER, 192, Wa1, INNER, ktb, INNER, INNER / 16, INNER);
  // 4) fused pairwise MLPs + online softmax + fc2 tail
  pt_attn_kernel<<<NT, 128, 0, stream>>>(
      r, qkv, qtb, ktb, Wt1, bt1, bntg, bntb, bntm, bntv, Wt2, bt2,
      Wa1, ba1, Wa2, ba2, W2, b2, bn2g, bn2b, bn2m, bn2v, skip, outp);
}